// CompressedSensingConvolutional_72971494359714
// MI455X (gfx1250) — compile-verified
//
#include <hip/hip_runtime.h>
#include <math.h>

// ---------------------------------------------------------------------------
// FISTA compressed-sensing solver for MI455X (gfx1250), persistent-workgroup.
//
//   inp : (64, 81, 3) f32      A : (81, 5184) f32     lam, mu : scalar f32
//   out : (64, 72, 72, 3) f32  == x after 200 FISTA iterations
//
// 192 independent (batch,channel) columns. 12 workgroups x 1024 threads
// (32 waves = 8 waves/SIMD for L2-latency hiding); each WG owns a 16-column
// tile and runs all 200 iterations on-chip (data L2-resident). Matmuls use
// V_WMMA_F32_16X16X4_F32 for full f32 accuracy. All zero-padding/re-packing
// of A happens once at init so the iteration loops are branch-free, and all
// per-wave loop bounds are readfirstlane-uniform (SGPR trip counts -> no
// EXEC-masked loops, unrolling + load pipelining enabled). Per-wave WMMA
// partials are dumped with ds_store_b128 and tree-reduced deterministically.
// ---------------------------------------------------------------------------

#define M_MEAS   81
#define DIM      5184
#define NCOL     192
#define TILE_N   16
#define NWG      12            // 192 / 16
#define THREADS  1024
#define WAVES    32
#define NITER    200
#define MT_TILES 6             // 81 rows padded to 96 = 6 x 16
#define D_TILES  324           // 5184 / 16
#define KSTEPS_A 1296          // 5184 / 4 (waves 0..15: 41 steps, 16..31: 40)
#define KSTEPS_B 21            // ceil(81/4): K-steps of the A^T GEMM
#define AY_ELTS  (MT_TILES * 256)       // 1536 = 96 x 16

#define TILE_ELTS (DIM * TILE_N)        // 82944 floats per 16-col tile
#define COLS_ELTS (DIM * NCOL)          // 995328
#define APAD_ROWS 96
#define APAD_ELTS (APAD_ROWS * DIM)     // 497664
#define A4_ELTS   (KSTEPS_B * DIM * 4)  // 435456

typedef __attribute__((ext_vector_type(2))) float v2f;   // A/B frag: 16x4 f32
typedef __attribute__((ext_vector_type(4))) float v4f;   // half C/D frag
typedef __attribute__((ext_vector_type(8))) float v8f;   // C/D frag: 16x16 f32

__device__ __forceinline__ v8f wmma4(v2f a, v2f b, v8f c) {
  // D = A(16x4) * B(4x16) + C(16x16), f32 everywhere.
  return __builtin_amdgcn_wmma_f32_16x16x4_f32(
      false, a, false, b, (short)0, c, false, false);
}

__device__ __forceinline__ v8f v8zero() {
  v8f z;
  #pragma unroll
  for (int i = 0; i < 8; ++i) z[i] = 0.0f;
  return z;
}

__global__ __launch_bounds__(THREADS, 1)
void fista_wmma_kernel(const float* __restrict__ inp,
                       const float* __restrict__ A,
                       const float* __restrict__ lam,
                       const float* __restrict__ mu,
                       float* __restrict__ out,
                       float* __restrict__ ws)
{
  // LDS: 32*1536*4 = 196608 B partials + 6144 B Ay + 5184 B inp (~203 KB).
  // P layout [wave][mt][half][n][8r]: each lane's 8 accumulator values are
  // contiguous -> 2x ds_store_b128 per tile.
  __shared__ float P[WAVES * AY_ELTS];
  __shared__ float Ay_s[AY_ELTS];                // Ay packed [mstep][n][4m]
  __shared__ float inp_s[M_MEAS * TILE_N];       // camera counts for tile

  const int tid  = threadIdx.x;
  const int lane = tid & 31;
  const int wave = tid >> 5;
  const int l15  = lane & 15;
  const int half = lane >> 4;          // 0 | 1
  const int koff = half << 1;          // 0 | 2 : K offset of frag VGPR0
  const int rhlf = half << 3;          // 0 | 8 : M offset of C/D frag rows
  const int wg   = blockIdx.x;
  const int colbase = wg * TILE_N;

  // Wave index as a compiler-provable SGPR value: gives uniform (s_cmp)
  // loops instead of EXEC-masked ones, re-enabling unroll + pipelining.
  const int wu = __builtin_amdgcn_readfirstlane(wave);
  const int ksA0 = wu * 40 + (wu < 16 ? wu : 16);  // contiguous K range
  const int nksA = 40 + (wu < 16 ? 1 : 0);         // 41/40 k-steps
  const int dt0  = wu * 10 + (wu < 4 ? wu : 4);    // contiguous d-tile range
  const int ndt  = 10 + (wu < 4 ? 1 : 0);          // 11/10 d-tiles

  // Workspace (L2-resident, ~16 MB total).
  // X/Y/Atb per-tile, packed [kstep][n][4k] so B-frags are one b64 load.
  float* AtbT = ws + (size_t)wg * TILE_ELTS;
  float* Xt   = ws + (size_t)COLS_ELTS     + (size_t)wg * TILE_ELTS;
  float* Yt   = ws + (size_t)2 * COLS_ELTS + (size_t)wg * TILE_ELTS;
  float* Apad = ws + (size_t)3 * COLS_ELTS;        // A rows padded 81->96
  float* A4   = Apad + APAD_ELTS;                  // A packed [ms][d][4m]

  const float mu_s = mu[0];
  const float thr  = lam[0] * mu_s;

  // ---- Init: stage this tile's camera counts into LDS ---------------------
  for (int i = tid; i < M_MEAS * TILE_N; i += THREADS) {
    int m = i >> 4, jj = i & 15;
    int j = colbase + jj;
    int b = j / 3, c = j - 3 * b;
    inp_s[i] = inp[(b * M_MEAS + m) * 3 + c];
  }
  __syncthreads();

  // ---- Init: Atb = A^T b (packed), zero X and Y ---------------------------
  for (int d = tid; d < DIM; d += THREADS) {
    float acc[TILE_N];
    #pragma unroll
    for (int jj = 0; jj < TILE_N; ++jj) acc[jj] = 0.0f;
    for (int m = 0; m < M_MEAS; ++m) {
      float a = A[m * DIM + d];
      #pragma unroll
      for (int jj = 0; jj < TILE_N; ++jj)
        acc[jj] = fmaf(a, inp_s[m * TILE_N + jj], acc[jj]);
    }
    int pb = (d >> 2) * 64 + (d & 3);
    #pragma unroll
    for (int jj = 0; jj < TILE_N; ++jj) {
      int idx = pb + jj * 4;
      AtbT[idx] = acc[jj];
      Xt[idx]   = 0.0f;
      Yt[idx]   = 0.0f;
    }
  }

  // ---- Init: zero-padded / re-packed copies of A (identical values from
  //      every WG -> benign write races; no grid sync available) -----------
  for (int m = 0; m < APAD_ROWS; ++m)
    for (int d = tid; d < DIM; d += THREADS)
      Apad[m * DIM + d] = (m < M_MEAS) ? A[m * DIM + d] : 0.0f;

  for (int s = 0; s < KSTEPS_B; ++s)
    for (int d = tid; d < DIM; d += THREADS) {
      #pragma unroll
      for (int mm = 0; mm < 4; ++mm) {
        int m = s * 4 + mm;
        A4[((size_t)s * DIM + d) * 4 + mm] =
            (m < M_MEAS) ? A[m * DIM + d] : 0.0f;
      }
    }
  __threadfence();
  __syncthreads();

  // ---- 200 FISTA iterations, entirely inside this workgroup --------------
  float t = 1.0f;
  for (int it = 0; it < NITER; ++it) {
    float t_new = (1.0f + sqrtf(1.0f + 4.0f * t * t)) * 0.5f;
    float coef  = (t - 1.0f) / t_new;

    // ==== Step A: Ay(96x16) = Apad * Ytile; contiguous per-wave K range ===
    v8f acc[MT_TILES];
    #pragma unroll
    for (int mt = 0; mt < MT_TILES; ++mt) acc[mt] = v8zero();

    #pragma unroll 2
    for (int s = 0; s < nksA; ++s) {
      int kk = ksA0 + s;
      v2f bf = *(const v2f*)(Yt + kk * 64 + l15 * 4 + koff);
      int k0 = kk * 4 + koff;
      #pragma unroll
      for (int mt = 0; mt < MT_TILES; ++mt) {
        v2f af = *(const v2f*)(Apad + (mt * 16 + l15) * DIM + k0);
        acc[mt] = wmma4(af, bf, acc[mt]);
      }
    }

    // Per-wave partials -> LDS with 2x ds_store_b128 per tile.
    {
      float* Pw = P + wave * AY_ELTS + half * 128 + l15 * 8;
      #pragma unroll
      for (int mt = 0; mt < MT_TILES; ++mt) {
        v4f lo = __builtin_shufflevector(acc[mt], acc[mt], 0, 1, 2, 3);
        v4f hi = __builtin_shufflevector(acc[mt], acc[mt], 4, 5, 6, 7);
        *(v4f*)(Pw + mt * 256)     = lo;
        *(v4f*)(Pw + mt * 256 + 4) = hi;
      }
    }
    __syncthreads();

    // Deterministic tree reduction (fixed wave order, no float atomics).
    for (int v = tid; v < AY_ELTS; v += THREADS) {
      int row = v >> 4, n = v & 15;
      int mt = row >> 4, rowl = row & 15;
      int fb = (mt * 2 + (rowl >> 3)) * 128 + n * 8 + (rowl & 7);
      float s = 0.0f;
      #pragma unroll
      for (int w = 0; w < WAVES; ++w) s += P[w * AY_ELTS + fb];
      Ay_s[(row >> 2) * 64 + n * 4 + (row & 3)] = s;   // packed [ms][n][4m]
    }
    __syncthreads();

    // ==== Step B: G = A^T * Ay (via A4), fused FISTA update ================
    for (int i = 0; i < ndt; ++i) {
      int dt   = dt0 + i;
      int d0   = dt * 16;
      int dcol = d0 + l15;                          // A^T frag row (M = d)
      v8f g = v8zero();

      #pragma unroll 7
      for (int s = 0; s < KSTEPS_B; ++s) {          // K padded to 84, no tail
        v2f af = *(const v2f*)(A4 + ((size_t)s * DIM + dcol) * 4 + koff);
        v2f bf = *(const v2f*)(Ay_s + s * 64 + l15 * 4 + koff);
        g = wmma4(af, bf, g);
      }

      // Fused gradient step + soft-threshold + momentum for 8 elems/lane.
      #pragma unroll
      for (int r = 0; r < 8; ++r) {
        int d   = d0 + r + rhlf;
        int idx = (d >> 2) * 64 + l15 * 4 + (d & 3);
        float yv = Yt[idx];
        float z  = fmaf(-mu_s, g[r] - AtbT[idx], yv);         // y - mu*grad
        float xn = copysignf(fmaxf(fabsf(z) - thr, 0.0f), z); // soft-thresh
        float xo = Xt[idx];
        float yn = fmaf(coef, xn - xo, xn);                   // momentum
        Xt[idx] = xn;
        Yt[idx] = yn;
      }
    }

    t = t_new;
    __threadfence();   // cross-CU visibility of X/Y within the WGP
    __syncthreads();
  }

  // ---- Write out: (B, 72, 72, 3) = (b, d, c) ------------------------------
  for (int d = tid; d < DIM; d += THREADS) {
    int pb = (d >> 2) * 64 + (d & 3);
    #pragma unroll
    for (int jj = 0; jj < TILE_N; ++jj) {
      int j = colbase + jj;
      int b = j / 3, c = j - 3 * b;
      out[(b * DIM + d) * 3 + c] = Xt[pb + jj * 4];
    }
  }
}

extern "C" void kernel_launch(void* const* d_in, const int* in_sizes, int n_in,
                              void* d_out, int out_size, void* d_ws, size_t ws_size,
                              hipStream_t stream) {
  (void)in_sizes; (void)n_in; (void)out_size; (void)ws_size;
  const float* inp = (const float*)d_in[0];   // (64, 81, 3)
  const float* A   = (const float*)d_in[1];   // (81, 5184)
  const float* lam = (const float*)d_in[2];   // (1,)
  const float* mu  = (const float*)d_in[3];   // (1,)
  float* out = (float*)d_out;                 // (64, 72, 72, 3)
  float* ws  = (float*)d_ws;                  // needs ~16 MB f32 scratch

  fista_wmma_kernel<<<dim3(NWG), dim3(THREADS), 0, stream>>>(
      inp, A, lam, mu, out, ws);
}